// MessageFunc_79276506349845
// MI455X (gfx1250) — compile-verified
//
#include <hip/hip_runtime.h>

typedef __attribute__((ext_vector_type(2))) float v2f;
typedef __attribute__((ext_vector_type(8))) float v8f;
typedef __attribute__((ext_vector_type(4))) int   v4i;

#define NODES    10000
#define NEIGH    20
#define EDGE_SZ  16
#define HID      128
#define OUT_SZ   32
#define XS       132   // LDS row stride for 128-wide rows (128+4: banks step 4/row)
#define W0S      18    // LDS row stride for 16-wide W0 rows (even -> b64 aligned)

#define AS1 __attribute__((address_space(1)))
#define AS3 __attribute__((address_space(3)))

#if defined(__has_builtin)
#  if __has_builtin(__builtin_amdgcn_global_load_async_to_lds_b128)
#    define HAVE_ASYNC_LDS 1
#  endif
#endif
#ifndef HAVE_ASYNC_LDS
#  define HAVE_ASYNC_LDS 0
#endif

static __device__ __forceinline__ v8f wmma_f32(v2f a, v2f b, v8f c) {
    // D = A(16x4) * B(4x16) + C(16x16), all fp32
    return __builtin_amdgcn_wmma_f32_16x16x4_f32(false, a, false, b, (short)0, c,
                                                 false, false);
}

#if HAVE_ASYNC_LDS
static __device__ __forceinline__ void async_copy_b128(const void* g, void* l) {
    // per-lane 16B: one instruction moves 512B straight into LDS (no VGPRs)
    __builtin_amdgcn_global_load_async_to_lds_b128((AS1 v4i*)g, (AS3 v4i*)l, 0, 0);
}
static __device__ __forceinline__ void wait_async0() {
#if defined(__has_builtin) && __has_builtin(__builtin_amdgcn_s_wait_asynccnt)
    __builtin_amdgcn_s_wait_asynccnt(0);
#else
    asm volatile("s_wait_asynccnt 0x0" ::: "memory");
#endif
}
#endif

// ---------------------------------------------------------------------------
// Kernel 1: A_v[v][o][k] = sum_i W2[32o+i][k] * h_w[v][i]   (per-o dense GEMM)
//           bh[v][o]    = sum_i b2[32o+i]    * h_w[v][i]
// grid = (NODES/16) * 32 blocks of 256 threads; wave w owns k-cols [16w,16w+16)
// ---------------------------------------------------------------------------
__global__ __launch_bounds__(256) void build_Av(
    const float* __restrict__ h_w,   // [NODES][32]
    const float* __restrict__ W2,    // [1024][128]
    const float* __restrict__ b2,    // [1024]
    float* __restrict__ A_ws,        // [NODES][32][128]
    float* __restrict__ bh_ws)       // [NODES][32]
{
    const int vt    = blockIdx.x >> 5;   // v-tile 0..624
    const int o     = blockIdx.x & 31;   // output-row slice 0..31
    const int tid   = threadIdx.x;
    const int wave  = tid >> 5;
    const int lane  = tid & 31;
    const int lhalf = lane >> 4;         // K-half select
    const int lmod  = lane & 15;         // M / N index

    const int nb    = wave * 16;         // k-column base
    const int vrowA = vt * 16 + lmod;

    v8f c = {};
#pragma unroll
    for (int kb = 0; kb < 32; kb += 4) {
        const int i = kb + 2 * lhalf;
        v2f a, b;
        a.x = h_w[vrowA * 32 + i];
        a.y = h_w[vrowA * 32 + i + 1];
        b.x = W2[(32 * o + i)     * HID + nb + lmod];
        b.y = W2[(32 * o + i + 1) * HID + nb + lmod];
        c = wmma_f32(a, b, c);
    }
#pragma unroll
    for (int r = 0; r < 8; ++r) {
        const int vrow = vt * 16 + 8 * lhalf + r;
        A_ws[((size_t)vrow * 32 + o) * HID + nb + lmod] = c[r];
    }

    if (tid < 16) {                       // bias fold: bh[v][o]
        const int v = vt * 16 + tid;
        float s = 0.f;
#pragma unroll
        for (int i = 0; i < 32; ++i) s += b2[32 * o + i] * h_w[v * 32 + i];
        bh_ws[v * 32 + o] = s;
    }
}

// ---------------------------------------------------------------------------
// Kernel 2: per-node fused MLP + contraction. 1 block / node, 8 wave32s.
//   x1 = relu(e @ W0^T + b0)   M=32(pad) N=128 K=16
//   x2 = relu(x1 @ W1^T + b1)  M=32      N=128 K=128
//   out = x2 @ A_v^T + bh      M=32      N=32  K=128   (rows >= 20 dropped)
// W1 (64KB) and A_v (16KB) are staged row-major via async B128 copies issued
// at kernel start and hidden behind the e-tile staging + the whole x1 layer.
// ---------------------------------------------------------------------------
__global__ __launch_bounds__(256) void fused_mlp(
    const float* __restrict__ e_vw,  // [E][16]
    const float* __restrict__ W0,    // [128][16]
    const float* __restrict__ b0,    // [128]
    const float* __restrict__ W1,    // [128][128]
    const float* __restrict__ b1,    // [128]
    const float* __restrict__ A_ws,  // [NODES][32][128]
    const float* __restrict__ bh_ws, // [NODES][32]
    float* __restrict__ out)         // [E][32]
{
    extern __shared__ float smem[];
    float* sA  = smem;               // 32*XS   A-operand activations (e, then x2)
    float* sX  = sA  + 32 * XS;      // 32*XS   x1 staging
    float* sW0 = sX  + 32 * XS;      // 128*W0S W0 row-major [n][k]
    float* sW1 = sW0 + 128 * W0S;    // 128*XS  W1 row-major [n][k]
    float* sAv = sW1 + 128 * XS;     // 32*XS   A_v row-major [o][k]
    float* sb0 = sAv + 32 * XS;      // 128
    float* sb1 = sb0 + 128;          // 128
    float* sbh = sb1 + 128;          // 32

    const int v     = blockIdx.x;
    const int tid   = threadIdx.x;
    const int wave  = tid >> 5;
    const int lane  = tid & 31;
    const int lhalf = lane >> 4;
    const int lmod  = lane & 15;
    const int nb    = wave * 16;

#if HAVE_ASYNC_LDS
    // ---- fire-and-forget staging: one async B128 per 512B row ----
    {
        const char* gW1 = (const char*)W1;
        const char* gAv = (const char*)(A_ws + (size_t)v * (OUT_SZ * HID));
        char*       lW1 = (char*)sW1;
        char*       lAv = (char*)sAv;
#pragma unroll
        for (int r = 0; r < 16; ++r) {       // W1: wave owns 16 rows
            const int n = wave * 16 + r;
            async_copy_b128(gW1 + (size_t)n * (HID * 4) + lane * 16,
                            lW1 + (size_t)n * (XS * 4) + lane * 16);
        }
#pragma unroll
        for (int r = 0; r < 4; ++r) {        // A_v: wave owns 4 rows
            const int o = wave * 4 + r;
            async_copy_b128(gAv + (size_t)o * (HID * 4) + lane * 16,
                            lAv + (size_t)o * (XS * 4) + lane * 16);
        }
    }
#else
    for (int idx = tid; idx < HID * HID; idx += 256) {     // sW1[n][k] = W1[n][k]
        const int n = idx >> 7, k = idx & 127;
        sW1[n * XS + k] = W1[idx];
    }
    for (int idx = tid; idx < OUT_SZ * HID; idx += 256) {  // sAv[o][k] = A_v[o][k]
        const int o = idx >> 7, k = idx & 127;
        sAv[o * XS + k] = A_ws[(size_t)v * (OUT_SZ * HID) + idx];
    }
#endif

    // ---- stage edge tile (rows 20..31 zero-padded) + W0 + biases ----
    for (int idx = tid; idx < 32 * EDGE_SZ; idx += 256) {
        const int row = idx >> 4, col = idx & 15;
        sA[row * XS + col] =
            (row < NEIGH) ? e_vw[((size_t)v * NEIGH + row) * EDGE_SZ + col] : 0.f;
    }
    for (int idx = tid; idx < HID * EDGE_SZ; idx += 256) { // sW0[n][k] = W0[n][k]
        const int n = idx >> 4, k = idx & 15;
        sW0[n * W0S + k] = W0[idx];
    }
    if (tid < 128) { sb0[tid] = b0[tid]; sb1[tid] = b1[tid]; }
    if (tid < 32)  { sbh[tid] = bh_ws[v * 32 + tid]; }
    __syncthreads();

    // ---- layer 0: x1 = relu(e @ W0^T + b0), wave owns N-cols [nb,nb+16) ----
    v8f c0 = {}, c1 = {};
#pragma unroll
    for (int kb = 0; kb < EDGE_SZ; kb += 4) {
        const int kk = kb + 2 * lhalf;
        v2f b;  b.x  = sW0[(nb + lmod) * W0S + kk];   b.y  = sW0[(nb + lmod) * W0S + kk + 1];
        v2f a0; a0.x = sA[lmod * XS + kk];            a0.y = sA[lmod * XS + kk + 1];
        v2f a1; a1.x = sA[(16 + lmod) * XS + kk];     a1.y = sA[(16 + lmod) * XS + kk + 1];
        c0 = wmma_f32(a0, b, c0);
        c1 = wmma_f32(a1, b, c1);
    }
#pragma unroll
    for (int r = 0; r < 8; ++r) {
        const int m = 8 * lhalf + r, n = nb + lmod;
        float t0 = c0[r] + sb0[n]; t0 = t0 > 0.f ? t0 : 0.f;
        float t1 = c1[r] + sb0[n]; t1 = t1 > 0.f ? t1 : 0.f;
        sX[m * XS + n]        = t0;
        sX[(16 + m) * XS + n] = t1;
    }
#if HAVE_ASYNC_LDS
    wait_async0();            // own wave's W1/A_v rows landed
#endif
    __syncthreads();          // everyone's rows landed + x1 visible

    // ---- layer 1: x2 = relu(x1 @ W1^T + b1) ----
    c0 = (v8f){}; c1 = (v8f){};
#pragma unroll
    for (int kb = 0; kb < HID; kb += 4) {
        const int kk = kb + 2 * lhalf;
        v2f b;  b.x  = sW1[(nb + lmod) * XS + kk];    b.y  = sW1[(nb + lmod) * XS + kk + 1];
        v2f a0; a0.x = sX[lmod * XS + kk];            a0.y = sX[lmod * XS + kk + 1];
        v2f a1; a1.x = sX[(16 + lmod) * XS + kk];     a1.y = sX[(16 + lmod) * XS + kk + 1];
        c0 = wmma_f32(a0, b, c0);
        c1 = wmma_f32(a1, b, c1);
    }
#pragma unroll
    for (int r = 0; r < 8; ++r) {
        const int m = 8 * lhalf + r, n = nb + lmod;
        float t0 = c0[r] + sb1[n]; t0 = t0 > 0.f ? t0 : 0.f;
        float t1 = c1[r] + sb1[n]; t1 = t1 > 0.f ? t1 : 0.f;
        sA[m * XS + n]        = t0;   // x2 overwrites edge tile
        sA[(16 + m) * XS + n] = t1;
    }
    __syncthreads();

    // ---- final: out = x2 @ A_v^T + bh (wave-uniform branch; EXEC stays full) --
    if (wave < 4) {
        const int mt = wave & 1;     // edge-row tile
        const int nt = wave >> 1;    // o tile
        v8f c = {};
#pragma unroll
        for (int kb = 0; kb < HID; kb += 4) {
            const int kk = kb + 2 * lhalf;
            v2f a; a.x = sA[(mt * 16 + lmod) * XS + kk];
                   a.y = sA[(mt * 16 + lmod) * XS + kk + 1];
            v2f b; b.x = sAv[(nt * 16 + lmod) * XS + kk];
                   b.y = sAv[(nt * 16 + lmod) * XS + kk + 1];
            c = wmma_f32(a, b, c);
        }
#pragma unroll
        for (int r = 0; r < 8; ++r) {
            const int row = mt * 16 + 8 * lhalf + r;
            if (row < NEIGH) {
                const int o = nt * 16 + lmod;
                out[((size_t)v * NEIGH + row) * OUT_SZ + o] = c[r] + sbh[o];
            }
        }
    }
}

// ---------------------------------------------------------------------------
extern "C" void kernel_launch(void* const* d_in, const int* in_sizes, int n_in,
                              void* d_out, int out_size, void* d_ws, size_t ws_size,
                              hipStream_t stream) {
    (void)in_sizes; (void)n_in; (void)out_size; (void)ws_size;
    // dict order: h_v, h_w, e_vw, W0, b0, W1, b1, W2, b2   (h_v unused)
    const float* h_w = (const float*)d_in[1];
    const float* e_vw = (const float*)d_in[2];
    const float* W0 = (const float*)d_in[3];
    const float* b0 = (const float*)d_in[4];
    const float* W1 = (const float*)d_in[5];
    const float* b1 = (const float*)d_in[6];
    const float* W2 = (const float*)d_in[7];
    const float* b2 = (const float*)d_in[8];
    float* out = (float*)d_out;

    float* A_ws  = (float*)d_ws;                          // NODES*32*128 f32 = 163.8 MB
    float* bh_ws = A_ws + (size_t)NODES * 32 * HID;       // NODES*32     f32 = 1.3 MB

    build_Av<<<dim3((NODES / 16) * 32), dim3(256), 0, stream>>>(
        h_w, W2, b2, A_ws, bh_ws);

    const size_t smem_floats = 32 * XS + 32 * XS + 128 * W0S + 128 * XS +
                               32 * XS + 128 + 128 + 32;
    fused_mlp<<<dim3(NODES), dim3(256), smem_floats * sizeof(float), stream>>>(
        e_vw, W0, b0, W1, b1, A_ws, bh_ws, out);
}